// SAGE_tune_21947282883085
// MI455X (gfx1250) — compile-verified
//
#include <hip/hip_runtime.h>

#define FDIM 128
#define MT 5  // row-tiles (16 rows each) per wave: 80 rows per block

typedef __attribute__((ext_vector_type(2))) float v2f;
typedef __attribute__((ext_vector_type(8))) float v8f;

// ---------------------------------------------------------------------------
// Zero a float buffer (agg + deg zeroed each layer; graph-capture safe).
// ---------------------------------------------------------------------------
__global__ void zero_f32(float* __restrict__ p, int n) {
  int i = blockIdx.x * blockDim.x + threadIdx.x;
  if (i < n) p[i] = 0.0f;
}

// ---------------------------------------------------------------------------
// Edge scatter: agg[dst] += (relu?)feat[src]; deg[dst] += 1.
// 32 threads per edge, 4 features each (float4 coalesced gather,
// 4 hardware global_atomic_add_f32 per thread).
// ---------------------------------------------------------------------------
template <int RELU>
__global__ void sage_scatter(const float* __restrict__ feat,
                             const int* __restrict__ src,
                             const int* __restrict__ dst,
                             float* __restrict__ agg,
                             float* __restrict__ deg, int E) {
  int t = blockIdx.x * blockDim.x + threadIdx.x;
  int e = t >> 5;
  if (e >= E) return;
  int c = (t & 31) << 2;  // feature offset: 0..124 step 4
  int s = src[e];
  int d = dst[e];
  const float4 v = *reinterpret_cast<const float4*>(feat + (size_t)s * FDIM + c);
  float x0 = v.x, x1 = v.y, x2 = v.z, x3 = v.w;
  if (RELU) {
    x0 = fmaxf(x0, 0.0f); x1 = fmaxf(x1, 0.0f);
    x2 = fmaxf(x2, 0.0f); x3 = fmaxf(x3, 0.0f);
  }
  float* ap = agg + (size_t)d * FDIM + c;
  __hip_atomic_fetch_add(ap + 0, x0, __ATOMIC_RELAXED, __HIP_MEMORY_SCOPE_AGENT);
  __hip_atomic_fetch_add(ap + 1, x1, __ATOMIC_RELAXED, __HIP_MEMORY_SCOPE_AGENT);
  __hip_atomic_fetch_add(ap + 2, x2, __ATOMIC_RELAXED, __HIP_MEMORY_SCOPE_AGENT);
  __hip_atomic_fetch_add(ap + 3, x3, __ATOMIC_RELAXED, __HIP_MEMORY_SCOPE_AGENT);
  if ((t & 31) == 0)
    __hip_atomic_fetch_add(deg + d, 1.0f, __ATOMIC_RELAXED, __HIP_MEMORY_SCOPE_AGENT);
}

// ---------------------------------------------------------------------------
// deg[i] -> 1 / max(deg[i], 1)
// ---------------------------------------------------------------------------
__global__ void rdeg_kernel(float* __restrict__ deg, int n) {
  int i = blockIdx.x * blockDim.x + threadIdx.x;
  if (i < n) deg[i] = 1.0f / fmaxf(deg[i], 1.0f);
}

// ---------------------------------------------------------------------------
// Fused: out = (agg * rdeg) @ Wl + bias + (RELU?)(xin) @ Wr
// V_WMMA_F32_16X16X4_F32; one wave32 owns MT 16x16 tiles in one column strip.
// Block = 256 threads = 8 waves -> 80 rows x 128 cols per block.
// B (4x16 weight slice) loaded once per K-step, reused by 2*MT WMMAs.
// Branch-free row clamp keeps EXEC all-1s around WMMA (no divergence).
//
// A layout (16x4 f32, MxK): lanes 0-15 -> M=lane, VGPR{0,1} = K{0,1};
//                           lanes 16-31 -> M=lane-16, VGPR{0,1} = K{2,3}.
// B layout (4x16 f32, KxN): lanes = N column; symmetric K striping.
// C/D layout (16x16 f32):   VGPR v -> M = v (+8 for upper half), N = lane&15.
// ---------------------------------------------------------------------------
template <int RELU>
__global__ void sage_gemm_wmma(const float* __restrict__ agg,
                               const float* __restrict__ rdeg,
                               const float* __restrict__ xin,
                               const float* __restrict__ Wl,
                               const float* __restrict__ Wr,
                               const float* __restrict__ bias,
                               float* __restrict__ out, int nrows) {
  const int lane = threadIdx.x & 31;
  const int wave = threadIdx.x >> 5;   // 0..7 -> column tile
  const int half = lane >> 4;          // 0: lanes 0-15, 1: lanes 16-31
  const int l    = lane & 15;
  const int row0 = blockIdx.x * (16 * MT);
  const int col0 = wave * 16;
  const int n    = col0 + l;

  const float* ap[MT];
  const float* xp[MT];
  float rd[MT];
#pragma unroll
  for (int t = 0; t < MT; ++t) {
    int r = row0 + t * 16 + l;
    r = (r < nrows) ? r : (nrows - 1);           // branch-free tail clamp
    rd[t] = rdeg[r];
    ap[t] = agg + (size_t)r * FDIM + half * 2;
    xp[t] = xin + (size_t)r * FDIM + half * 2;
  }
  const float* wlp = Wl + (size_t)(half * 2) * FDIM + n;
  const float* wrp = Wr + (size_t)(half * 2) * FDIM + n;

  v8f acc[MT] = {};
  for (int k = 0; k < FDIM; k += 4) {
    v2f bL, bR;
    bL.x = wlp[(size_t)k * FDIM];
    bL.y = wlp[(size_t)(k + 1) * FDIM];
    bR.x = wrp[(size_t)k * FDIM];
    bR.y = wrp[(size_t)(k + 1) * FDIM];

    v2f aL[MT], aR[MT];
#pragma unroll
    for (int t = 0; t < MT; ++t) {
      v2f m = *reinterpret_cast<const v2f*>(ap[t] + k);
      aL[t].x = m.x * rd[t];
      aL[t].y = m.y * rd[t];
      v2f xv = *reinterpret_cast<const v2f*>(xp[t] + k);
      if (RELU) { xv.x = fmaxf(xv.x, 0.0f); xv.y = fmaxf(xv.y, 0.0f); }
      aR[t] = xv;
    }
    // Two passes so same-accumulator WMMAs sit MT instructions apart
    // (hides WMMA->WMMA RAW hazard with independent work, not NOPs).
#pragma unroll
    for (int t = 0; t < MT; ++t)
      acc[t] = __builtin_amdgcn_wmma_f32_16x16x4_f32(
          false, aL[t], false, bL, (short)0, acc[t], false, false);
#pragma unroll
    for (int t = 0; t < MT; ++t)
      acc[t] = __builtin_amdgcn_wmma_f32_16x16x4_f32(
          false, aR[t], false, bR, (short)0, acc[t], false, false);
  }

  const float bv = bias[n];
#pragma unroll
  for (int t = 0; t < MT; ++t) {
#pragma unroll
    for (int v = 0; v < 8; ++v) {
      const int m = row0 + t * 16 + v + half * 8;
      if (m < nrows) out[(size_t)m * FDIM + n] = acc[t][v] + bv;
    }
  }
}

// ---------------------------------------------------------------------------
// Host-side orchestration (graph-capture safe: only async launches on stream)
// Inputs: 0=x 1=src 2=dst, then (Wl,Wr,b) x 3 layers.
// Outputs: d_out = [h2 (layer2 pre-relu), h3] each N*128 f32.
// Workspace: agg[N*128] | deg[N] | h1[N*128]
// ---------------------------------------------------------------------------
extern "C" void kernel_launch(void* const* d_in, const int* in_sizes, int n_in,
                              void* d_out, int out_size, void* d_ws, size_t ws_size,
                              hipStream_t stream) {
  const float* x   = (const float*)d_in[0];
  const int*   src = (const int*)d_in[1];
  const int*   dst = (const int*)d_in[2];
  const float* Wl[3] = {(const float*)d_in[3], (const float*)d_in[6], (const float*)d_in[9]};
  const float* Wr[3] = {(const float*)d_in[4], (const float*)d_in[7], (const float*)d_in[10]};
  const float* bs[3] = {(const float*)d_in[5], (const float*)d_in[8], (const float*)d_in[11]};

  const int N = in_sizes[0] / FDIM;
  const int E = in_sizes[1];

  float* agg = (float*)d_ws;
  float* deg = agg + (size_t)N * FDIM;
  float* h1  = deg + N;
  float* h2  = (float*)d_out;
  float* h3  = h2 + (size_t)N * FDIM;

  const float* lin[3]  = {x, h1, h2};
  float*       lout[3] = {h1, h2, h3};

  const dim3 blk(256);
  const int zeroN = N * FDIM + N;  // agg+deg contiguous
  const unsigned zgrid = (unsigned)((zeroN + 255) / 256);
  const long long sthreads = (long long)E * 32;
  const unsigned sgrid = (unsigned)((sthreads + 255) / 256);
  const unsigned rgrid = (unsigned)((N + 255) / 256);
  const unsigned ggrid = (unsigned)((N + 16 * MT - 1) / (16 * MT));  // 1250 for N=100000

  for (int L = 0; L < 3; ++L) {
    zero_f32<<<zgrid, blk, 0, stream>>>(agg, zeroN);
    if (L == 0)
      sage_scatter<0><<<sgrid, blk, 0, stream>>>(lin[L], src, dst, agg, deg, E);
    else
      sage_scatter<1><<<sgrid, blk, 0, stream>>>(lin[L], src, dst, agg, deg, E);
    rdeg_kernel<<<rgrid, blk, 0, stream>>>(deg, N);
    if (L == 0)
      sage_gemm_wmma<0><<<ggrid, blk, 0, stream>>>(agg, deg, lin[L],
                                                   Wl[L], Wr[L], bs[L], lout[L], N);
    else
      sage_gemm_wmma<1><<<ggrid, blk, 0, stream>>>(agg, deg, lin[L],
                                                   Wl[L], Wr[L], bs[L], lout[L], N);
  }
}